// RelPositionMultiHeadedAttention_45148696216760
// MI455X (gfx1250) — compile-verified
//
#include <hip/hip_runtime.h>
#include <hip/hip_bf16.h>

typedef __attribute__((ext_vector_type(2))) float v2f;
typedef __attribute__((ext_vector_type(8))) float v8f;

#define TOKENS   32768
#define EMB      128
#define HEADS    8
#define SEQ      4096          // TOKENS / HEADS
#define N3E      384           // 3*EMB

#define KSTRIDE  132           // K tile LDS row stride (floats): b64 frag reads conflict-free
#define VSTRIDE  136           // V tile LDS row stride (floats): b32 frag reads conflict-free
#define PSTRIDE  20            // P exchange row stride (floats)

// scale = 1/sqrt(128); fold log2(e) so softmax can use exp2 (v_exp_f32 trans op)
#define QSCALE   (0.08838834764831845f * 1.4426950408889634f)

// ---------------------------------------------------------------------------
// Kernel 1: fused QKV projection + bias + head-interleave scatter.
// One wave computes one 16x16 tile of  qkv[32768,384] = X[32768,128] @ W[128,384]
// with 32x v_wmma_f32_16x16x4_f32, then scatters to per-head q/k/v layouts,
// adding b_qkv and u_bias/v_bias, pre-folding scale*log2(e) into q.
// ---------------------------------------------------------------------------
__global__ __launch_bounds__(32)
void qkv_proj_kernel(const float* __restrict__ X,     // [T, 128]
                     const float* __restrict__ W,     // [128, 384]
                     const float* __restrict__ bqkv,  // [384]
                     const float* __restrict__ ubias, // [8, 128]
                     const float* __restrict__ vbias, // [8, 128]
                     float* __restrict__ qws,         // [H, S, E]  (pre-scaled)
                     float* __restrict__ kws,         // [H, S, E]
                     float* __restrict__ vws)         // [H, S, E]
{
    const int lane = threadIdx.x & 31;
    const int half = lane >> 4;
    const int lcol = lane & 15;
    const int k0   = half * 2;
    const int mt   = blockIdx.x;      // 0..2047  (M tile)
    const int nt   = blockIdx.y;      // 0..23    (N tile)

    const float* arow = X + (size_t)(mt * 16 + lcol) * EMB;
    const int    bcol = nt * 16 + lcol;

    v8f acc = {};
#pragma unroll
    for (int kk = 0; kk < 32; ++kk) {
        const int kb = kk * 4 + k0;
        v2f a = *(const v2f*)(arow + kb);
        v2f b;
        b.x = W[(size_t)kb * N3E + bcol];
        b.y = W[(size_t)(kb + 1) * N3E + bcol];
        acc = __builtin_amdgcn_wmma_f32_16x16x4_f32(
            false, a, false, b, (short)0, acc, false, false);
    }

    const int part = (nt * 16) >> 7;           // 0=q, 1=k, 2=v
    const int e    = nt * 16 + lcol - part * EMB;
    const float bias = bqkv[nt * 16 + lcol];

#pragma unroll
    for (int r = 0; r < 8; ++r) {
        const int t = mt * 16 + r + half * 8;   // token index
        const int h = t & (HEADS - 1);
        const int s = t >> 3;
        const size_t idx = ((size_t)h * SEQ + s) * EMB + e;
        const float v = acc[r] + bias;
        if (part == 0)      qws[idx] = (v + ubias[h * EMB + e]) * QSCALE;
        else if (part == 1) kws[idx] =  v + vbias[h * EMB + e];
        else                vws[idx] =  v;
    }
}

// ---------------------------------------------------------------------------
// Async global->LDS copy of one 16x128 f32 tile (8 KB) using the CDNA5
// async-copy path (GLOBAL_LOAD_ASYNC_TO_LDS_B128, tracked by ASYNCcnt).
// 512 16-byte chunks spread over 256 threads (2 per thread).
// ---------------------------------------------------------------------------
__device__ __forceinline__
void async_copy_tile(const float* __restrict__ ghead, int j, float* lds_tile,
                     int strideW, int tid)
{
#pragma unroll
    for (int c = tid; c < 512; c += 256) {
        const int row = c >> 5;        // 0..15
        const int col = c & 31;        // 16B chunk within row
        const float* g = ghead + (size_t)(j * 16 + row) * EMB + col * 4;
        const unsigned ldsOff = (unsigned)(size_t)(lds_tile + row * strideW + col * 4);
        asm volatile("global_load_async_to_lds_b128 %0, %1, off"
                     :: "v"(ldsOff), "v"((unsigned long long)(size_t)g)
                     : "memory");
    }
}

// ---------------------------------------------------------------------------
// Kernel 2: flash attention. 256-thread blocks (8 waves); wave w owns a 16-row
// q strip of a 128-row block. K/V 16x128 tiles are staged in LDS (double
// buffered) via async copies so all 8 waves share each tile (8x less L2
// traffic). Per key tile: 32x wmma (Q*K^T), online softmax (shfl_xor row
// reductions), P relaid C->A layout through padded per-wave LDS, 32x wmma
// (P*V).
// ---------------------------------------------------------------------------
__global__ __launch_bounds__(256)
void flash_attn_kernel(const float* __restrict__ qws,
                       const float* __restrict__ kws,
                       const float* __restrict__ vws,
                       float* __restrict__ out)      // [H, S, E]
{
    __shared__ float kbuf[2][16 * KSTRIDE];   // 2 x 8448 B
    __shared__ float vbuf[2][16 * VSTRIDE];   // 2 x 8704 B
    __shared__ float pbuf[8][16 * PSTRIDE];   // 8 x 1280 B

    const int tid  = threadIdx.x;
    const int wid  = tid >> 5;
    const int lane = tid & 31;
    const int half = lane >> 4;
    const int lcol = lane & 15;
    const int k0   = half * 2;
    const int h    = blockIdx.y;
    const int qbase = blockIdx.x * 128 + wid * 16;   // this wave's q rows

    const float* Q = qws + (size_t)h * SEQ * EMB;
    const float* K = kws + (size_t)h * SEQ * EMB;
    const float* V = vws + (size_t)h * SEQ * EMB;

    // Resident Q A-fragments (64 VGPRs)
    v2f aq[32];
    {
        const float* qrow = Q + (size_t)(qbase + lcol) * EMB;
#pragma unroll
        for (int kk = 0; kk < 32; ++kk)
            aq[kk] = *(const v2f*)(qrow + kk * 4 + k0);
    }

    v8f acc[8];
#pragma unroll
    for (int et = 0; et < 8; ++et) acc[et] = (v8f){};
    float mrow[8], lrow[8];
#pragma unroll
    for (int r = 0; r < 8; ++r) { mrow[r] = -1e30f; lrow[r] = 0.0f; }

    // Prologue: stage tile 0 into buffer 0
    async_copy_tile(K, 0, kbuf[0], KSTRIDE, tid);
    async_copy_tile(V, 0, vbuf[0], VSTRIDE, tid);

    float* pw = pbuf[wid];

#pragma unroll 1
    for (int j = 0; j < SEQ / 16; ++j) {
        const int cur = j & 1;
        asm volatile("s_wait_asynccnt 0" ::: "memory");   // own async copies done
        __syncthreads();                                  // tile j visible to all waves

        if (j + 1 < SEQ / 16) {                           // prefetch tile j+1
            async_copy_tile(K, j + 1, kbuf[cur ^ 1], KSTRIDE, tid);
            async_copy_tile(V, j + 1, vbuf[cur ^ 1], VSTRIDE, tid);
        }

        // ---- S = Q * K_j^T ; B frag from LDS: K[lcol][kk*4+k0 + {0,1}]
        const float* krow = &kbuf[cur][lcol * KSTRIDE];
        v8f s = {};
#pragma unroll
        for (int kk = 0; kk < 32; ++kk) {
            v2f b = *(const v2f*)(krow + kk * 4 + k0);
            s = __builtin_amdgcn_wmma_f32_16x16x4_f32(
                false, aq[kk], false, b, (short)0, s, false, false);
        }

        // ---- online softmax (log2 domain; q pre-scaled by scale*log2e)
#pragma unroll
        for (int r = 0; r < 8; ++r) {
            float sv = s[r];
            float rmax = sv;
            rmax = fmaxf(rmax, __shfl_xor(rmax, 1, 32));
            rmax = fmaxf(rmax, __shfl_xor(rmax, 2, 32));
            rmax = fmaxf(rmax, __shfl_xor(rmax, 4, 32));
            rmax = fmaxf(rmax, __shfl_xor(rmax, 8, 32));
            const float mnew = fmaxf(mrow[r], rmax);
            const float p    = exp2f(sv - mnew);
            float rsum = p;
            rsum += __shfl_xor(rsum, 1, 32);
            rsum += __shfl_xor(rsum, 2, 32);
            rsum += __shfl_xor(rsum, 4, 32);
            rsum += __shfl_xor(rsum, 8, 32);
            const float corr = exp2f(mrow[r] - mnew);
            lrow[r] = lrow[r] * corr + rsum;
            mrow[r] = mnew;
            s[r] = p;
#pragma unroll
            for (int et = 0; et < 8; ++et) acc[et][r] *= corr;
        }

        // ---- relayout P: C-layout -> per-wave LDS -> A-layout (wave-sync)
        asm volatile("" ::: "memory");
#pragma unroll
        for (int r = 0; r < 8; ++r)
            pw[(r + half * 8) * PSTRIDE + lcol] = s[r];
        asm volatile("s_wait_dscnt 0" ::: "memory");
        v2f ap[4];
#pragma unroll
        for (int t = 0; t < 4; ++t)
            ap[t] = *(const v2f*)(&pw[lcol * PSTRIDE + t * 4 + k0]);
        asm volatile("" ::: "memory");

        // ---- O += P * V_j ; B frag from LDS: V[t*4+k0 + {0,1}][et*16+lcol]
#pragma unroll
        for (int t = 0; t < 4; ++t) {
            const float* vr0 = &vbuf[cur][(t * 4 + k0) * VSTRIDE];
            const float* vr1 = vr0 + VSTRIDE;
#pragma unroll
            for (int et = 0; et < 8; ++et) {
                v2f bv;
                bv.x = vr0[et * 16 + lcol];
                bv.y = vr1[et * 16 + lcol];
                acc[et] = __builtin_amdgcn_wmma_f32_16x16x4_f32(
                    false, ap[t], false, bv, (short)0, acc[et], false, false);
            }
        }
        __syncthreads();   // all waves done with tile j before its buffer is reused
    }

    // ---- finalize: out = acc / l
#pragma unroll
    for (int r = 0; r < 8; ++r) {
        const float inv = 1.0f / lrow[r];
        const size_t row = (size_t)h * SEQ + (size_t)(qbase + r + half * 8);
#pragma unroll
        for (int et = 0; et < 8; ++et)
            out[row * EMB + et * 16 + lcol] = acc[et][r] * inv;
    }
}

// ---------------------------------------------------------------------------
extern "C" void kernel_launch(void* const* d_in, const int* in_sizes, int n_in,
                              void* d_out, int out_size, void* d_ws, size_t ws_size,
                              hipStream_t stream) {
    const float* query  = (const float*)d_in[0];  // [T, E]
    const float* W_qkv  = (const float*)d_in[3];  // [E, 3E]
    const float* b_qkv  = (const float*)d_in[4];  // [3E]
    const float* u_bias = (const float*)d_in[5];  // [H, E]
    const float* v_bias = (const float*)d_in[6];  // [H, E]
    float* out = (float*)d_out;                   // [H, S, E]

    float* qws = (float*)d_ws;                    // 16 MB each
    float* kws = qws + (size_t)TOKENS * EMB;
    float* vws = kws + (size_t)TOKENS * EMB;

    dim3 g1(TOKENS / 16, N3E / 16);               // 2048 x 24 tiles
    qkv_proj_kernel<<<g1, 32, 0, stream>>>(query, W_qkv, b_qkv, u_bias, v_bias,
                                           qws, kws, vws);

    dim3 g2(SEQ / 128, HEADS);                    // 32 x 8 blocks, 256 thr each
    flash_attn_kernel<<<g2, 256, 0, stream>>>(qws, kws, vws, out);
}